// XingLoss_59519656788220
// MI455X (gfx1250) — compile-verified
//
#include <hip/hip_runtime.h>
#include <stdint.h>

// ---------------------------------------------------------------------------
// XingLoss on MI455X (gfx1250): streaming reduction over (8192, 3072, 2) fp32.
// Memory-bound (201 MB -> scalar, ~8.6us at 23.3 TB/s). Data path: TDM
// tensor_load_to_lds stages each 24KB row into LDS (double-buffered,
// TENSORcnt-synchronized), waves compute crossing terms from LDS,
// deterministic two-stage reduction (no atomics).
// ---------------------------------------------------------------------------

typedef unsigned int v4u __attribute__((ext_vector_type(4)));
typedef int          v8i __attribute__((ext_vector_type(8)));
typedef int          v4i __attribute__((ext_vector_type(4)));

#define P_ROWS        8192
#define ROW_FLOATS    6144      /* 3072 points * 2 floats */
#define TRIPLES       1024      /* 3072 / 3 */
#define NBLOCKS       1024
#define ROWS_PER_BLK  (P_ROWS / NBLOCKS)
#define NTHREADS      256
#define TOTAL_TRIPLES 8388608.0f

// Issue one TDM DMA: copy ROW_FLOATS fp32 from gsrc (global) to LDS offset.
// D# layout per CDNA5 ISA ch.8 (group0 128b, group1 256b; groups 2/3 unused
// for a <=2D tensor and passed as zeros). This toolchain's builtin takes a
// 5th zero-filled int32x8 group before the cache-policy immediate.
__device__ __forceinline__ void tdm_load_row(const float* gsrc, uint32_t lds_byte_off) {
    uint64_t ga = (uint64_t)(uintptr_t)gsrc;
    v4u g0;
    g0.x = 1u;                                   // count=1 (valid), user mode
    g0.y = lds_byte_off;                         // lds_addr (bytes)
    g0.z = (uint32_t)ga;                         // global_addr[31:0]
    g0.w = (uint32_t)((ga >> 32) & 0x01FFFFFFu)  // global_addr[56:32]
         | 0x80000000u;                          // type=2 ("image") in [127:126]
    v8i g1;
    g1[0] = (int)(2u << 16);                             // data_size=2 -> 4 bytes
    g1[1] = (int)(((uint32_t)ROW_FLOATS & 0xFFFFu) << 16); // tensor_dim0 lo16
    g1[2] = (int)(((uint32_t)ROW_FLOATS >> 16) | (1u << 16)); // dim0 hi16 | tensor_dim1=1
    g1[3] = (int)((uint32_t)ROW_FLOATS << 16);           // tile_dim0 = 6144 in [127:112]
    g1[4] = 0;                                           // tile_dim1=0, tile_dim2=0
    g1[5] = ROW_FLOATS;                                  // tensor_dim0_stride lo32
    g1[6] = 0;                                           // stride0 hi16 | stride1 lo16
    g1[7] = 0;                                           // stride1 hi32
    v4i z4 = {0, 0, 0, 0};
    v8i z8 = {0, 0, 0, 0, 0, 0, 0, 0};
    __builtin_amdgcn_tensor_load_to_lds(g0, g1, z4, z4, z8, 0); // TENSORcnt++
}

__global__ __launch_bounds__(NTHREADS)
void xing_rows_kernel(const float* __restrict__ pts, float* __restrict__ partial) {
    __shared__ float buf[2][ROW_FLOATS];   // 48 KB double buffer
    __shared__ float wsum[NTHREADS / 32];

    const int tid = threadIdx.x;
    const int bid = blockIdx.x;
    const size_t row0 = (size_t)bid * ROWS_PER_BLK;

    // Prologue: wave 0 kicks off DMA of the first row.
    if (tid < 32) {
        tdm_load_row(pts + row0 * ROW_FLOATS, (uint32_t)(size_t)&buf[0][0]);
    }

    float acc = 0.0f;
    for (int i = 0; i < ROWS_PER_BLK; ++i) {
        if (tid < 32) {
            if (i + 1 < ROWS_PER_BLK) {
                // Prefetch next row into the other buffer, then wait until
                // only that one DMA is outstanding (row i has landed).
                tdm_load_row(pts + (row0 + i + 1) * ROW_FLOATS,
                             (uint32_t)(size_t)&buf[(i + 1) & 1][0]);
                __builtin_amdgcn_s_wait_tensorcnt(1);
            } else {
                __builtin_amdgcn_s_wait_tensorcnt(0);
            }
        }
        __syncthreads();   // row i visible to all waves

        const float* row = buf[i & 1];
#pragma unroll
        for (int j0 = 0; j0 < TRIPLES; j0 += NTHREADS) {
            const int j = j0 + tid;
            const int b = 6 * j;
            const float p1x = row[b + 0], p1y = row[b + 1];
            const float p2x = row[b + 2], p2y = row[b + 3];
            const float p3x = row[b + 4], p3y = row[b + 5];
            const int b4 = (j == TRIPLES - 1) ? 0 : b + 6;   // roll(p1,-1) wraps in-row
            const float p4x = row[b4], p4y = row[b4 + 1];

            const float v12x = p2x - p1x, v12y = p2y - p1y;
            const float v23x = p3x - p2x, v23y = p3y - p2y;
            const float v43x = p4x - p3x, v43y = p4y - p3y;

            const float c1 = v12x * v23y - v12y * v23x;          // D1 predicate
            const float c2 = v12x * v43y - v12y * v43x;
            const float den = sqrtf((v12x * v12x + v12y * v12y) *
                                    (v43x * v43x + v43y * v43y));
            const float d2 = c2 / den;
            acc += (c1 > 0.0f) ? fmaxf(-d2, 0.0f) : fmaxf(d2, 0.0f);
        }
        __syncthreads();   // everyone done reading buf[i&1] before it is re-DMA'd
    }

    // wave32 reduction, then across the 8 waves (deterministic order).
    for (int off = 16; off > 0; off >>= 1)
        acc += __shfl_down(acc, off, 32);
    if ((tid & 31) == 0) wsum[tid >> 5] = acc;
    __syncthreads();
    if (tid == 0) {
        float s = 0.0f;
        for (int w = 0; w < NTHREADS / 32; ++w) s += wsum[w];
        partial[bid] = s;
    }
}

__global__ __launch_bounds__(256)
void xing_final_kernel(const float* __restrict__ partial, float* __restrict__ out) {
    __shared__ float wsum[8];
    const int tid = threadIdx.x;
    float s = partial[tid] + partial[tid + 256] + partial[tid + 512] + partial[tid + 768];
    for (int off = 16; off > 0; off >>= 1)
        s += __shfl_down(s, off, 32);
    if ((tid & 31) == 0) wsum[tid >> 5] = s;
    __syncthreads();
    if (tid == 0) {
        float t = 0.0f;
        for (int w = 0; w < 8; ++w) t += wsum[w];
        out[0] = t * (1.0f / TOTAL_TRIPLES);
    }
}

extern "C" void kernel_launch(void* const* d_in, const int* in_sizes, int n_in,
                              void* d_out, int out_size, void* d_ws, size_t ws_size,
                              hipStream_t stream) {
    (void)in_sizes; (void)n_in; (void)out_size; (void)ws_size;
    const float* pts = (const float*)d_in[0];     // (8192, 3072, 2) fp32
    float* partial   = (float*)d_ws;              // 1024 floats of scratch
    float* out       = (float*)d_out;             // scalar fp32

    xing_rows_kernel<<<NBLOCKS, NTHREADS, 0, stream>>>(pts, partial);
    xing_final_kernel<<<1, 256, 0, stream>>>(partial, out);
}